// UnifiedWorldModel_60619168416508
// MI455X (gfx1250) — compile-verified
//
#include <hip/hip_runtime.h>
#include <math.h>

typedef float v2f __attribute__((ext_vector_type(2)));
typedef float v8f __attribute__((ext_vector_type(8)));
typedef unsigned int u32x4 __attribute__((ext_vector_type(4)));
typedef int i32x4 __attribute__((ext_vector_type(4)));
typedef int i32x8 __attribute__((ext_vector_type(8)));

#define D_    32
#define D2_   64
#define OUTF_ 128
#define BATCH_ 1024
#define TM_   64            // batch rows per workgroup
#define NT_   128           // 4 waves * 32 lanes
#define DT_C  0.01f
#define MAXS_ 0.1f

__device__ __forceinline__ float wave_sum(float x) {
#pragma unroll
  for (int m = 16; m > 0; m >>= 1) x += __shfl_xor(x, m, 32);
  return x;
}

// single-instruction v_rcp_f32 (~1 ulp) instead of the IEEE div expansion
__device__ __forceinline__ float fast_rcp(float x) {
  return __builtin_amdgcn_rcpf(x);
}

// branch-free tanh: -2|x| never overflows exp, so no range fixup needed
__device__ __forceinline__ float fast_tanh(float x) {
  const float t = __expf(-2.f * fabsf(x));   // (0,1]
  return copysignf((1.f - t) * fast_rcp(1.f + t), x);
}

__device__ __forceinline__ float fast_silu(float x) {
  return x * fast_rcp(1.f + __expf(-x));
}

__device__ __forceinline__ v8f wmma4(v2f a, v2f b, v8f c) {
  // D = A(16x4,f32) * B(4x16,f32) + C(16x16,f32)
  return __builtin_amdgcn_wmma_f32_16x16x4_f32(false, a, false, b, (short)0, c,
                                               false, false);
}

// Issue one TDM 2D tile load Global->LDS with optional LDS row padding.
//   width_elts : dwords per row (tile_dim0 == tensor_dim0)
//   rows       : rows in tile   (tile_dim1 == tensor_dim1)
//   stride_elts: global row stride in dwords (tensor_dim0_stride)
//   pi_code    : pad_interval code (pad after 2^(pi+1) dwords)
//   pa_code    : pad_amount code  (pa+1 dwords of pad)
__device__ __forceinline__ void tdm_load_2d(const void* gptr, unsigned lds_off,
                                            unsigned width_elts, unsigned rows,
                                            unsigned stride_elts,
                                            unsigned pi_code, unsigned pa_code,
                                            bool pad_en) {
  const unsigned long long ga = (unsigned long long)(uintptr_t)gptr;
  u32x4 g0;
  g0.x = 1u;                                            // count=1 (valid D#)
  g0.y = lds_off;                                       // lds_addr (bytes)
  g0.z = (unsigned)(ga & 0xFFFFFFFFu);                  // global_addr[31:0]
  g0.w = ((unsigned)(ga >> 32) & 0x01FFFFFFu)           // global_addr[56:32]
         | (2u << 30);                                  // type = 2 ("image")
  i32x8 g1;
  g1[0] = (int)((2u << 16)                              // data_size = 4B
                | ((pad_en ? 1u : 0u) << 20)            // pad_enable
                | (pi_code << 22) | (pa_code << 25));   // pad geometry
  g1[1] = (int)((width_elts & 0xFFFFu) << 16);          // tensor_dim0 lo16
  g1[2] = (int)(((width_elts >> 16) & 0xFFFFu)          // tensor_dim0 hi16
                | ((rows & 0xFFFFu) << 16));            // tensor_dim1 lo16
  g1[3] = (int)(((rows >> 16) & 0xFFFFu)                // tensor_dim1 hi16
                | ((width_elts & 0xFFFFu) << 16));      // tile_dim0
  g1[4] = (int)(rows & 0xFFFFu);                        // tile_dim1 (tile_dim2=0)
  g1[5] = (int)stride_elts;                             // tensor_dim0_stride lo32
  g1[6] = 0;                                            // stride0 hi / stride1 lo
  g1[7] = 0;
  const i32x4 zz4 = {0, 0, 0, 0};
  const i32x8 zz8 = {0, 0, 0, 0, 0, 0, 0, 0};
  // amdgpu-toolchain (clang-23) 6-arg form: (g0, g1, g2, g3, gx, cpol)
  __builtin_amdgcn_tensor_load_to_lds(g0, g1, zz4, zz4, zz8, 0);
}

__global__ __launch_bounds__(NT_, 1)
void uwm_fused(const float* __restrict__ z0, const float* __restrict__ v0,
               const float* __restrict__ Lambda,
               const float* __restrict__ log_alpha,
               const float* __restrict__ scale,
               const float* __restrict__ Wf, const float* __restrict__ bf,
               const float* __restrict__ W1, const float* __restrict__ b1,
               const float* __restrict__ W2, const float* __restrict__ b2,
               const int* __restrict__ n_steps_p, float* __restrict__ out) {
  // LDS: per-wave-private row slabs; weights staged once via TDM.
  constexpr int ZLD  = 72;   // [z | v] row stride (64 cols + 8 pad)
  constexpr int TLD  = 72;   // temp (F then decoder-hidden) row stride
  constexpr int WFLD = 65;
  constexpr int W1LD = 33;
  __shared__ float zv [TM_][ZLD];
  __shared__ float tmp[TM_][TLD];
  __shared__ float sWf[D_ ][WFLD];   // Wf[n][k], n<32, k<64
  __shared__ float sW1[D2_][W1LD];   // W1[n][k], n<64, k<32

  const int tid  = threadIdx.x;
  const int lane = tid & 31;
  const int wave = tid >> 5;
  const int half = lane >> 4;        // K half-select for WMMA fragments
  const int ml   = lane & 15;        // M (A) / N (B,C) lane coordinate
  const int mrow = wave * 16;        // wave-owned 16-row slab
  const int gbase = blockIdx.x * TM_;

  // Pull the streamed W2 (32KB) into L2 early.
  __builtin_prefetch(&W2[(2 * tid + 0) * 32], 0, 0);
  __builtin_prefetch(&W2[(2 * tid + 1) * 32], 0, 0);

  const int   nsteps = *n_steps_p;
  const float alpha  = expf(log_alpha[0]);
  const float bscale = fminf(fabsf(scale[0]), MAXS_);
  const float lam_c  = fmaxf(Lambda[lane], 1e-6f);  // clipped eigenvalues

  // ---- TDM staging: Global -> LDS with hardware row padding --------------
  // zv z-half : 64 rows x 32 dw, pad 40 dw after every 32 dw  -> stride 72
  // zv v-half : same geometry, LDS base + 128B (interleaves [z|v])
  // sWf       : 32 rows x 64 dw, pad 1 dw after every 64 dw   -> stride 65
  // sW1       : 64 rows x 32 dw, pad 1 dw after every 32 dw   -> stride 33
  if (wave == 0) {
    tdm_load_2d(z0 + (size_t)gbase * D_, (unsigned)(uintptr_t)&zv[0][0],
                32u, (unsigned)TM_, 32u, /*pi=*/4u, /*pa=*/39u, true);
    tdm_load_2d(v0 + (size_t)gbase * D_, (unsigned)(uintptr_t)&zv[0][D_],
                32u, (unsigned)TM_, 32u, /*pi=*/4u, /*pa=*/39u, true);
    tdm_load_2d(Wf, (unsigned)(uintptr_t)&sWf[0][0],
                64u, (unsigned)D_, 64u, /*pi=*/5u, /*pa=*/0u, true);
    tdm_load_2d(W1, (unsigned)(uintptr_t)&sW1[0][0],
                32u, (unsigned)D2_, 32u, /*pi=*/4u, /*pa=*/0u, true);
    __builtin_amdgcn_s_wait_tensorcnt(0);
  }
  __syncthreads();

  for (int s = 0; s < nsteps; ++s) {
    // ---- F = tanh([z|v] @ Wf^T + bf) * bscale : M=16/wave, N=32, K=64 ----
#pragma unroll
    for (int nt = 0; nt < 2; ++nt) {
      v8f acc = {0.f, 0.f, 0.f, 0.f, 0.f, 0.f, 0.f, 0.f};
#pragma unroll
      for (int ks = 0; ks < 16; ++ks) {
        const int k0 = ks * 4 + 2 * half;
        v2f a, b;
        a.x = zv[mrow + ml][k0];        a.y = zv[mrow + ml][k0 + 1];
        b.x = sWf[nt * 16 + ml][k0];    b.y = sWf[nt * 16 + ml][k0 + 1];
        acc = wmma4(a, b, acc);
      }
      const int col = nt * 16 + ml;
      const float bias = bf[col];
#pragma unroll
      for (int r = 0; r < 8; ++r) {
        const int row = mrow + r + 8 * half;
        tmp[row][col] = fast_tanh(acc[r] + bias) * bscale;
      }
    }

    // ---- geodesic step: exact Christoffel contraction + Sherman-Morrison.
    // metric quadratic in z  =>  central FD is exact =>
    //   Gamma^k_ij v^i v^j = g^{kl} (2a (z.v) v_l + a |v|^2 z_l)
    // g + eps I = diag(Lam + a|z|^2 + eps) + 2a z z^T  (diag + rank-1)
#pragma unroll 1
    for (int r0 = 0; r0 < 16; ++r0) {
      const int row = mrow + r0;
      const float z = zv[row][lane];
      const float v = zv[row][lane + D_];
      const float F = tmp[row][lane];
      const float s_zz = wave_sum(z * z);
      const float s_zv = wave_sum(z * v);
      const float s_vv = wave_sum(v * v);
      const float d   = lam_c + alpha * s_zz + 1e-6f;  // diag of g + eps I
      const float rd  = fast_rcp(d);
      const float u   = 2.f * alpha * s_zv * v + alpha * s_vv * z;
      const float q   = (F - u) * rd;                  // a = g^{-1}(F - u)
      const float zd  = z * rd;
      const float t1  = wave_sum(z * q);
      const float t2  = wave_sum(z * zd);
      const float coef = 2.f * alpha * t1 * fast_rcp(1.f + 2.f * alpha * t2);
      const float a   = q - coef * zd;
      const float vn  = v + DT_C * a;                  // semi-implicit Euler
      const float zn  = z + DT_C * vn;
      zv[row][lane]       = zn;
      zv[row][lane + D_]  = vn;
    }

    // ---- decoder L1: h = SiLU(z_new @ W1^T + b1) : N=64, K=32 ----
#pragma unroll
    for (int nt = 0; nt < 4; ++nt) {
      v8f acc = {0.f, 0.f, 0.f, 0.f, 0.f, 0.f, 0.f, 0.f};
#pragma unroll
      for (int ks = 0; ks < 8; ++ks) {
        const int k0 = ks * 4 + 2 * half;
        v2f a, b;
        a.x = zv[mrow + ml][k0];        a.y = zv[mrow + ml][k0 + 1];
        b.x = sW1[nt * 16 + ml][k0];    b.y = sW1[nt * 16 + ml][k0 + 1];
        acc = wmma4(a, b, acc);
      }
      const int col = nt * 16 + ml;
      const float bias = b1[col];
#pragma unroll
      for (int r = 0; r < 8; ++r) {
        const int row = mrow + r + 8 * half;
        tmp[row][col] = fast_silu(acc[r] + bias);
      }
    }

    // ---- decoder L2: y = h @ W2^T + b2 : N=128, K=64 (B from L2$) ----
    float* outp = out + ((size_t)s * BATCH_ + gbase) * OUTF_;
#pragma unroll 2
    for (int nt = 0; nt < 8; ++nt) {
      v8f acc = {0.f, 0.f, 0.f, 0.f, 0.f, 0.f, 0.f, 0.f};
      const int col = nt * 16 + ml;
#pragma unroll
      for (int ks = 0; ks < 16; ++ks) {
        const int k0 = ks * 4 + 2 * half;
        v2f a, b;
        a.x = tmp[mrow + ml][k0];       a.y = tmp[mrow + ml][k0 + 1];
        b.x = W2[col * D2_ + k0];       b.y = W2[col * D2_ + k0 + 1];
        acc = wmma4(a, b, acc);
      }
      const float bias = b2[col];
#pragma unroll
      for (int r = 0; r < 8; ++r) {
        const int row = mrow + r + 8 * half;
        outp[(size_t)row * OUTF_ + col] = acc[r] + bias;
      }
    }
  }
}

extern "C" void kernel_launch(void* const* d_in, const int* in_sizes, int n_in,
                              void* d_out, int out_size, void* d_ws,
                              size_t ws_size, hipStream_t stream) {
  const float* z0        = (const float*)d_in[0];
  const float* v0        = (const float*)d_in[1];
  const float* Lambda    = (const float*)d_in[2];
  const float* log_alpha = (const float*)d_in[3];
  const float* scale     = (const float*)d_in[4];
  const float* Wf        = (const float*)d_in[5];
  const float* bf        = (const float*)d_in[6];
  const float* W1        = (const float*)d_in[7];
  const float* b1        = (const float*)d_in[8];
  const float* W2        = (const float*)d_in[9];
  const float* b2        = (const float*)d_in[10];
  const int*   nsteps    = (const int*)d_in[11];
  float* out = (float*)d_out;

  dim3 grid(BATCH_ / TM_);   // 16 workgroups
  dim3 block(NT_);           // 128 threads = 4 waves
  hipLaunchKernelGGL(uwm_fused, grid, block, 0, stream,
                     z0, v0, Lambda, log_alpha, scale, Wf, bf, W1, b1, W2, b2,
                     nsteps, out);
}